// MultiTaskLoss_10591389352207
// MI455X (gfx1250) — compile-verified
//
#include <hip/hip_runtime.h>
#include <math.h>

// ---------------------------------------------------------------------------
// MultiTaskLoss for MI455X (gfx1250, wave32).
//
// Cost model: reading album_logits (131072 x 1000 f32 = 524 MB) once at
// 23.3 TB/s ~= 22.5 us dominates everything. Strategy:
//   K2 row_lse : per-row logsumexp. Pass 1: coalesced float4 max scan (HBM).
//                Pass 2: exp + row-sum via V_WMMA_F32_16X16X4_F32 with
//                B = ones (matrix pipe does the reduction in exact f32);
//                re-reads the same 64KB/wave -> L2 hits, HBM stays ~524 MB.
//   K3 pairs   : sorted-COO segment walk (deterministic, no atomics):
//                per_pair = lse[s] - logits[s,a]; per-song sum/count.
//   K4/K5      : fixed-tree block reductions -> scalar loss.
// ---------------------------------------------------------------------------

typedef __attribute__((ext_vector_type(2))) float v2f;
typedef __attribute__((ext_vector_type(8))) float v8f;

#if defined(__has_builtin)
#if __has_builtin(__builtin_amdgcn_wmma_f32_16x16x4_f32)
#define HAVE_WMMA_F32X4 1
#endif
#endif
#ifndef HAVE_WMMA_F32X4
#define HAVE_WMMA_F32X4 0
#endif

#define ROWS_PER_WAVE 16
#define WAVES_PER_BLOCK 8
#define ROWS_PER_BLOCK (ROWS_PER_WAVE * WAVES_PER_BLOCK)

// ------------------------------ K1: init ws --------------------------------
__global__ void mtl_init_kernel(float* __restrict__ songSum,
                                int* __restrict__ songCnt, int B) {
  int i = blockIdx.x * blockDim.x + threadIdx.x;
  if (i < B) {
    songSum[i] = 0.0f;
    songCnt[i] = 0;
  }
}

// --------------------- K2: per-row logsumexp via WMMA ----------------------
__global__ __launch_bounds__(256) void mtl_row_lse_kernel(
    const float* __restrict__ logits, float* __restrict__ lse, int B, int A) {
  const int lane = threadIdx.x & 31;
  const int w = threadIdx.x >> 5;
  const int rowBase = blockIdx.x * ROWS_PER_BLOCK + w * ROWS_PER_WAVE;
  const bool active = (rowBase + ROWS_PER_WAVE) <= B;  // wave-uniform
  const int A4 = A >> 2;
  const int cMax = A4 << 2;

  __shared__ float sums[WAVES_PER_BLOCK][16];

  // laneRowMax: lane l ends holding max of row (rowBase + (l & 15)).
  float laneRowMax = 0.0f;

  if (active) {
    // ---- Pass 1: coalesced row max (float4, lane-strided) ----
    for (int r = 0; r < ROWS_PER_WAVE; ++r) {
      const float* rp = logits + (size_t)(rowBase + r) * (size_t)A;
      const float4* rp4 = (const float4*)rp;
      float m = -INFINITY;
      for (int i = lane; i < A4; i += 32) {
        float4 v = rp4[i];
        m = fmaxf(m, fmaxf(fmaxf(v.x, v.y), fmaxf(v.z, v.w)));
      }
      for (int i = cMax + lane; i < A; i += 32) m = fmaxf(m, rp[i]);
#pragma unroll
      for (int k = 16; k >= 1; k >>= 1) m = fmaxf(m, __shfl_xor(m, k, 32));
      laneRowMax = ((lane & 15) == r) ? m : laneRowMax;
    }
  }

  // ---- Pass 2: sum(exp(x - max)) over 16 rows using the matrix pipe ----
  // A-matrix f32 16x4 layout: lanes 0-15 feed K=0,1 ; lanes 16-31 feed K=2,3.
  // With B = ones(4x16), D accumulates exact-f32 row sums in every column.
  v8f acc = {0.f, 0.f, 0.f, 0.f, 0.f, 0.f, 0.f, 0.f};
#if !HAVE_WMMA_F32X4
  float fsum = 0.0f;  // fallback: per-lane partial, reduced with shfl_xor(16)
#endif
  if (active) {
    const float* rp = logits + (size_t)(rowBase + (lane & 15)) * (size_t)A +
                      ((lane >> 4) << 1);
#if HAVE_WMMA_F32X4
    v2f ones;
    ones[0] = 1.0f;
    ones[1] = 1.0f;
#endif
    for (int c = 0; c < cMax; c += 4) {
      float2 x = *(const float2*)(rp + c);
      float e0 = __expf(x.x - laneRowMax);
      float e1 = __expf(x.y - laneRowMax);
#if HAVE_WMMA_F32X4
      v2f a;
      a[0] = e0;
      a[1] = e1;
      acc = __builtin_amdgcn_wmma_f32_16x16x4_f32(
          false, a, false, ones, (short)0, acc, false, false);
#else
      fsum += e0 + e1;
#endif
    }
  }

#if HAVE_WMMA_F32X4
  // C/D 16x16 f32 layout: vgpr j holds row j (lanes 0-15) / row j+8 (16-31).
  if (active) {
    if (lane == 0) {
      sums[w][0] = acc[0]; sums[w][1] = acc[1]; sums[w][2] = acc[2];
      sums[w][3] = acc[3]; sums[w][4] = acc[4]; sums[w][5] = acc[5];
      sums[w][6] = acc[6]; sums[w][7] = acc[7];
    } else if (lane == 16) {
      sums[w][8]  = acc[0]; sums[w][9]  = acc[1]; sums[w][10] = acc[2];
      sums[w][11] = acc[3]; sums[w][12] = acc[4]; sums[w][13] = acc[5];
      sums[w][14] = acc[6]; sums[w][15] = acc[7];
    }
  }
  __syncthreads();
  if (active && lane < 16) {
    int row = rowBase + lane;
    float s = sums[w][lane];
    for (int cc = cMax; cc < A; ++cc)  // A%4 tail (none for A=1000)
      s += __expf(logits[(size_t)row * (size_t)A + cc] - laneRowMax);
    lse[row] = laneRowMax + __logf(s);
  }
#else
  __syncthreads();
  if (active) {
    float s = fsum + __shfl_xor(fsum, 16, 32);  // combine column halves
    if (lane < 16) {
      int row = rowBase + lane;
      for (int cc = cMax; cc < A; ++cc)
        s += __expf(logits[(size_t)row * (size_t)A + cc] - laneRowMax);
      lse[row] = laneRowMax + __logf(s);
    }
  }
#endif
}

// ------------- K3: sorted-COO segment walk (deterministic) -----------------
__global__ void mtl_pair_kernel(const float* __restrict__ logits,
                                const float* __restrict__ lse,
                                const int* __restrict__ song,
                                const int* __restrict__ album,
                                float* __restrict__ songSum,
                                int* __restrict__ songCnt, int A, int T) {
  int t = blockIdx.x * blockDim.x + threadIdx.x;
  if (t >= T) return;
  int s = song[t];
  if (t > 0 && song[t - 1] == s) return;  // not a segment head
  const float l = lse[s];
  const float* row = logits + (size_t)s * (size_t)A;
  float acc = 0.0f;
  int cnt = 0;
  int j = t;
  do {
    acc += l - row[album[j]];  // -log p = lse - logit
    ++cnt;
    ++j;
  } while (j < T && song[j] == s);
  songSum[s] = acc;
  songCnt[s] = cnt;
}

// ---------------- K4: per-block partial sums (fixed tree) ------------------
__global__ __launch_bounds__(256) void mtl_partial_kernel(
    const float* __restrict__ preds, const float* __restrict__ targets,
    const float* __restrict__ songSum, const int* __restrict__ songCnt,
    float4* __restrict__ partials, int B) {
  __shared__ float4 red[256];
  int i = blockIdx.x * 256 + threadIdx.x;
  float sse = 0.f, nv = 0.f, sm = 0.f, ns = 0.f;
  if (i < B) {
    float tg = targets[i];
    float d = preds[i] - tg;
    if (tg >= 0.0f) { sse = d * d; nv = 1.0f; }
    int c = songCnt[i];
    if (c > 0) { sm = songSum[i] / (float)c; ns = 1.0f; }
  }
  red[threadIdx.x] = make_float4(sse, nv, sm, ns);
  __syncthreads();
  for (int off = 128; off > 0; off >>= 1) {
    if (threadIdx.x < off) {
      float4 a = red[threadIdx.x], b = red[threadIdx.x + off];
      red[threadIdx.x] = make_float4(a.x + b.x, a.y + b.y, a.z + b.z, a.w + b.w);
    }
    __syncthreads();
  }
  if (threadIdx.x == 0) partials[blockIdx.x] = red[0];
}

// ---------------------- K5: final scalar (1 block) -------------------------
__global__ __launch_bounds__(256) void mtl_final_kernel(
    const float4* __restrict__ partials, int n, float* __restrict__ out) {
  __shared__ float4 red[256];
  float4 a = make_float4(0.f, 0.f, 0.f, 0.f);
  for (int i = threadIdx.x; i < n; i += 256) {
    float4 p = partials[i];
    a.x += p.x; a.y += p.y; a.z += p.z; a.w += p.w;
  }
  red[threadIdx.x] = a;
  __syncthreads();
  for (int off = 128; off > 0; off >>= 1) {
    if (threadIdx.x < off) {
      float4 u = red[threadIdx.x], v = red[threadIdx.x + off];
      red[threadIdx.x] = make_float4(u.x + v.x, u.y + v.y, u.z + v.z, u.w + v.w);
    }
    __syncthreads();
  }
  if (threadIdx.x == 0) {
    float4 r = red[0];
    float rating = (r.y > 0.f) ? r.x / fmaxf(r.y, 1.0f) : 0.0f;
    float album  = (r.w > 0.f) ? r.z / fmaxf(r.w, 1.0f) : 0.0f;
    out[0] = 1.0f * rating + 0.5f * album;
  }
}

// ------------------------------ launcher -----------------------------------
extern "C" void kernel_launch(void* const* d_in, const int* in_sizes, int n_in,
                              void* d_out, int out_size, void* d_ws,
                              size_t ws_size, hipStream_t stream) {
  const float* preds   = (const float*)d_in[0];  // (B,1)
  const float* targets = (const float*)d_in[1];  // (B,)
  const float* logits  = (const float*)d_in[2];  // (B,A)
  const int* song      = (const int*)d_in[3];    // (T,) sorted
  const int* album     = (const int*)d_in[4];    // (T,)

  const int B = in_sizes[1];
  const int A = in_sizes[2] / B;
  const int T = in_sizes[3];

  // Workspace layout (floats): lse[B] | songSum[B] | songCnt[B] | partials
  float* ws = (float*)d_ws;
  float* lse = ws;
  float* songSum = ws + B;
  int* songCnt = (int*)(ws + 2 * (size_t)B);
  const int nPart = (B + 255) / 256;
  float4* partials = (float4*)(ws + 3 * (size_t)B);

  mtl_init_kernel<<<(B + 255) / 256, 256, 0, stream>>>(songSum, songCnt, B);
  mtl_row_lse_kernel<<<(B + ROWS_PER_BLOCK - 1) / ROWS_PER_BLOCK, 256, 0,
                       stream>>>(logits, lse, B, A);
  mtl_pair_kernel<<<(T + 255) / 256, 256, 0, stream>>>(logits, lse, song,
                                                       album, songSum, songCnt,
                                                       A, T);
  mtl_partial_kernel<<<nPart, 256, 0, stream>>>(preds, targets, songSum,
                                                songCnt, partials, B);
  mtl_final_kernel<<<1, 256, 0, stream>>>(partials, nPart, (float*)d_out);
}